// GATNet_24369644437899
// MI455X (gfx1250) — compile-verified
//
#include <hip/hip_runtime.h>
#include <math.h>

#define IN_F   128
#define NHEADS 4
#define LRELU  0.2f
#define LN_EPS 1e-5f

typedef __attribute__((ext_vector_type(2))) float v2f;
typedef __attribute__((ext_vector_type(8))) float v8f;

// ---------------------------------------------------------------------------
// GEMM: Bo[N x 128] = A[N x 128] * W[128 x 128], fp32 via V_WMMA_F32_16X16X4_F32
// Block = 256 threads (8 waves). Each block: 16-row strip; each wave: one
// 16x16 output tile (8 column tiles cover 128 cols). K=128 -> 32 chained WMMAs.
// ---------------------------------------------------------------------------
__global__ __launch_bounds__(256)
void gemm128_wmma(const float* __restrict__ A, const float* __restrict__ W,
                  float* __restrict__ Bo, int N) {
  __shared__ float As[16 * 132];           // padded stride vs 64 banks
  const int row0 = blockIdx.x * 16;
  const int tid  = threadIdx.x;
  const int wave = tid >> 5;
  const int lane = tid & 31;

  // cooperative, coalesced load of the 16x128 A tile
  #pragma unroll
  for (int i = 0; i < 8; ++i) {
    int idx = tid + i * 256;
    int r = idx >> 7, c = idx & 127;
    int rr = row0 + r; if (rr >= N) rr = N - 1;
    As[r * 132 + c] = A[(size_t)rr * IN_F + c];
  }
  __syncthreads();

  const int n0    = wave * 16;             // this wave's output column tile
  const int lhalf = lane & 15;
  const int khalf = (lane >> 4) * 2;       // 0 for lanes 0-15, 2 for 16-31
  v8f acc = {};

  #pragma unroll
  for (int s = 0; s < 32; ++s) {
    const int k0 = s * 4;
    v2f a, b;
    // A frag: lane<16 holds (M=lane, K=k0,k0+1); lane>=16 holds (M=lane-16, K=k0+2,k0+3)
    a.x = As[lhalf * 132 + k0 + khalf];
    a.y = As[lhalf * 132 + k0 + khalf + 1];
    // B frag: VGPR0 rows K=k0/k0+2 striped over lanes, VGPR1 rows K=k0+1/k0+3
    b.x = W[(size_t)(k0 + khalf)     * IN_F + n0 + lhalf];
    b.y = W[(size_t)(k0 + khalf + 1) * IN_F + n0 + lhalf];
    acc = __builtin_amdgcn_wmma_f32_16x16x4_f32(false, a, false, b,
                                                (short)0, acc, false, false);
  }

  // C/D layout: VGPR r -> (M=r, N=lane) lanes 0-15 ; (M=8+r, N=lane-16) lanes 16-31
  const int mbase = row0 + ((lane >> 4) ? 8 : 0);
  #pragma unroll
  for (int r = 0; r < 8; ++r) {
    int m = mbase + r;
    if (m < N) Bo[(size_t)m * IN_F + n0 + lhalf] = acc[r];
  }
}

// ---------------------------------------------------------------------------
// Attention coefficients: a_s[n,h] = sum_c h[n,h,c]*att_src[h,c] (wave32 reduce)
// ---------------------------------------------------------------------------
__global__ __launch_bounds__(128)
void att_coeff(const float* __restrict__ h, const float* __restrict__ asrc,
               const float* __restrict__ adst, float* __restrict__ a_s,
               float* __restrict__ a_d, int heads) {
  const int n = blockIdx.x;
  const int t = threadIdx.x;               // 128 threads = 4 waves
  float v = h[(size_t)n * IN_F + t];
  float s = v * asrc[t];
  float d = v * adst[t];
  #pragma unroll
  for (int off = 16; off > 0; off >>= 1) {
    s += __shfl_down(s, off, 32);
    d += __shfl_down(d, off, 32);
  }
  __shared__ float ss[4], sd[4];
  const int wave = t >> 5, lane = t & 31;
  if (heads == NHEADS) {                   // head == wave (HID==32, wave32)
    if (lane == 0) {
      a_s[(size_t)n * NHEADS + wave] = s;
      a_d[(size_t)n * NHEADS + wave] = d;
    }
  } else {                                 // heads == 1: combine the 4 waves
    if (lane == 0) { ss[wave] = s; sd[wave] = d; }
    __syncthreads();
    if (t == 0) {
      a_s[n] = ss[0] + ss[1] + ss[2] + ss[3];
      a_d[n] = sd[0] + sd[1] + sd[2] + sd[3];
    }
  }
}

// ---------------------------------------------------------------------------
__global__ void fill_f32(float* __restrict__ p, float v, size_t n) {
  size_t i = (size_t)blockIdx.x * blockDim.x + threadIdx.x;
  if (i < n) p[i] = v;
}

__device__ __forceinline__ void edge_endpoints(const int* __restrict__ ei,
                                               int e, int E, int& s, int& d) {
  if (e < E) { s = ei[e]; d = ei[E + e]; }       // edge_index rows: [src..][dst..]
  else       { s = e - E; d = s; }               // appended self-loops
}

__device__ __forceinline__ void atomicMaxF(float* addr, float val) {
  if (val >= 0.0f) atomicMax((int*)addr, __float_as_int(val));
  else             atomicMin((unsigned int*)addr, __float_as_uint(val));
}

__device__ __forceinline__ float lrelu(float v) {
  return v >= 0.0f ? v : LRELU * v;
}

// pass 1: per-destination running max of leaky-relu'd logits
__global__ void edge_max_k(const int* __restrict__ ei, const float* __restrict__ a_s,
                           const float* __restrict__ a_d, float* __restrict__ emax,
                           int E, int N, int heads) {
  int i = blockIdx.x * blockDim.x + threadIdx.x;
  int total = (E + N) * heads;
  if (i >= total) return;
  int e = i / heads, hd = i - e * heads;
  int s, d; edge_endpoints(ei, e, E, s, d);
  float v = lrelu(a_s[(size_t)s * heads + hd] + a_d[(size_t)d * heads + hd]);
  atomicMaxF(&emax[(size_t)d * heads + hd], v);
}

// pass 2: softmax denominator
__global__ void edge_sum_k(const int* __restrict__ ei, const float* __restrict__ a_s,
                           const float* __restrict__ a_d, const float* __restrict__ emax,
                           float* __restrict__ den, int E, int N, int heads) {
  int i = blockIdx.x * blockDim.x + threadIdx.x;
  int total = (E + N) * heads;
  if (i >= total) return;
  int e = i / heads, hd = i - e * heads;
  int s, d; edge_endpoints(ei, e, E, s, d);
  float v = lrelu(a_s[(size_t)s * heads + hd] + a_d[(size_t)d * heads + hd]);
  float ex = expf(v - emax[(size_t)d * heads + hd]);
  atomicAdd(&den[(size_t)d * heads + hd], ex);
}

// pass 3: out[dst] += alpha * h[src]; 32 lanes per edge, float4 per lane
__global__ __launch_bounds__(256)
void edge_scatter_k(const int* __restrict__ ei, const float* __restrict__ a_s,
                    const float* __restrict__ a_d, const float* __restrict__ emax,
                    const float* __restrict__ den, const float* __restrict__ h,
                    float* __restrict__ out, int E, int N, int heads) {
  int e = blockIdx.x * 8 + (threadIdx.x >> 5);
  if (e >= E + N) return;
  const int lane = threadIdx.x & 31;
  int s, d; edge_endpoints(ei, e, E, s, d);
  const int hd = (heads == NHEADS) ? (lane >> 3) : 0;   // 32 channels/head
  float v = lrelu(a_s[(size_t)s * heads + hd] + a_d[(size_t)d * heads + hd]);
  float alpha = expf(v - emax[(size_t)d * heads + hd]) /
                (den[(size_t)d * heads + hd] + 1e-16f);
  const float4 hv = *(const float4*)(h + (size_t)s * IN_F + lane * 4);
  float* op = out + (size_t)d * IN_F + lane * 4;
  atomicAdd(op + 0, hv.x * alpha);
  atomicAdd(op + 1, hv.y * alpha);
  atomicAdd(op + 2, hv.z * alpha);
  atomicAdd(op + 3, hv.w * alpha);
}

// ---------------------------------------------------------------------------
// bias + LayerNorm(128) + PReLU + residual   (blocks 0 and 1)
// ---------------------------------------------------------------------------
__global__ __launch_bounds__(128)
void post_block(const float* __restrict__ agg, const float* __restrict__ bias,
                const float* __restrict__ g, const float* __restrict__ be,
                const float* __restrict__ p, const float* __restrict__ resid,
                float* __restrict__ dst) {
  const int n = blockIdx.x, t = threadIdx.x;
  float v = agg[(size_t)n * IN_F + t] + bias[t];
  float s = v, s2 = v * v;
  #pragma unroll
  for (int off = 16; off > 0; off >>= 1) {
    s  += __shfl_down(s,  off, 32);
    s2 += __shfl_down(s2, off, 32);
  }
  __shared__ float w1[4], w2[4];
  const int wave = t >> 5, lane = t & 31;
  if (lane == 0) { w1[wave] = s; w2[wave] = s2; }
  __syncthreads();
  const float tot  = w1[0] + w1[1] + w1[2] + w1[3];
  const float tot2 = w2[0] + w2[1] + w2[2] + w2[3];
  const float mean = tot * (1.0f / IN_F);
  const float var  = tot2 * (1.0f / IN_F) - mean * mean;
  float y = (v - mean) * rsqrtf(var + LN_EPS) * g[t] + be[t];
  const float pw = p[0];
  y = y >= 0.0f ? y : pw * y;
  dst[(size_t)n * IN_F + t] = y + resid[(size_t)n * IN_F + t];
}

// block 2: out = agg + bias + resid
__global__ void final_add(const float* __restrict__ agg, const float* __restrict__ bias,
                          const float* __restrict__ resid, float* __restrict__ out,
                          int total) {
  int i = blockIdx.x * blockDim.x + threadIdx.x;
  if (i >= total) return;
  out[i] = agg[i] + bias[i & (IN_F - 1)] + resid[i];
}

// ---------------------------------------------------------------------------
static void gat_layer(const float* feat, const float* W, const float* asr,
                      const float* adt, const int* ei, int N, int E, int heads,
                      float* Bm, float* C, float* a_s, float* a_d,
                      float* emax, float* den, hipStream_t stream) {
  gemm128_wmma<<<(N + 15) / 16, 256, 0, stream>>>(feat, W, Bm, N);
  att_coeff<<<N, 128, 0, stream>>>(Bm, asr, adt, a_s, a_d, heads);
  const size_t nh = (size_t)N * heads;
  fill_f32<<<(unsigned)((nh + 255) / 256), 256, 0, stream>>>(emax, -INFINITY, nh);
  fill_f32<<<(unsigned)((nh + 255) / 256), 256, 0, stream>>>(den, 0.0f, nh);
  const size_t nc = (size_t)N * IN_F;
  fill_f32<<<(unsigned)((nc + 255) / 256), 256, 0, stream>>>(C, 0.0f, nc);
  const int tot = (E + N) * heads;
  edge_max_k<<<(tot + 255) / 256, 256, 0, stream>>>(ei, a_s, a_d, emax, E, N, heads);
  edge_sum_k<<<(tot + 255) / 256, 256, 0, stream>>>(ei, a_s, a_d, emax, den, E, N, heads);
  edge_scatter_k<<<(E + N + 7) / 8, 256, 0, stream>>>(ei, a_s, a_d, emax, den, Bm, C,
                                                      E, N, heads);
}

extern "C" void kernel_launch(void* const* d_in, const int* in_sizes, int n_in,
                              void* d_out, int out_size, void* d_ws, size_t ws_size,
                              hipStream_t stream) {
  const float* x   = (const float*)d_in[0];
  const int*   ei  = (const int*)  d_in[1];
  const float* W0  = (const float*)d_in[2];
  const float* as0 = (const float*)d_in[3];
  const float* ad0 = (const float*)d_in[4];
  const float* b0  = (const float*)d_in[5];
  const float* g0  = (const float*)d_in[6];
  const float* be0 = (const float*)d_in[7];
  const float* p0  = (const float*)d_in[8];
  const float* W1  = (const float*)d_in[9];
  const float* as1 = (const float*)d_in[10];
  const float* ad1 = (const float*)d_in[11];
  const float* b1  = (const float*)d_in[12];
  const float* g1  = (const float*)d_in[13];
  const float* be1 = (const float*)d_in[14];
  const float* p1  = (const float*)d_in[15];
  const float* W2  = (const float*)d_in[16];
  const float* as2 = (const float*)d_in[17];
  const float* ad2 = (const float*)d_in[18];
  const float* b2  = (const float*)d_in[19];

  const int N = in_sizes[0] / IN_F;
  const int E = in_sizes[1] / 2;

  // workspace layout (floats): A | Bm | C | a_s | a_d | emax | den
  float* ws  = (float*)d_ws;
  float* A   = ws;                                  // layer output / residual
  float* Bm  = A   + (size_t)N * IN_F;              // GEMM result h
  float* C   = Bm  + (size_t)N * IN_F;              // aggregation target
  float* a_s = C   + (size_t)N * IN_F;
  float* a_d = a_s + (size_t)N * NHEADS;
  float* emx = a_d + (size_t)N * NHEADS;
  float* den = emx + (size_t)N * NHEADS;

  // ---- block 0: GAT(128->4x32) + LN + PReLU + residual(x) -> A
  gat_layer(x, W0, as0, ad0, ei, N, E, NHEADS, Bm, C, a_s, a_d, emx, den, stream);
  post_block<<<N, 128, 0, stream>>>(C, b0, g0, be0, p0, x, A);

  // ---- block 1: GAT(128->4x32) + LN + PReLU + residual(A) -> A (in-place safe)
  gat_layer(A, W1, as1, ad1, ei, N, E, NHEADS, Bm, C, a_s, a_d, emx, den, stream);
  post_block<<<N, 128, 0, stream>>>(C, b1, g1, be1, p1, A, A);

  // ---- block 2: GAT(128->128, heads=1) + bias + residual(A) -> d_out
  gat_layer(A, W2, as2, ad2, ei, N, E, 1, Bm, C, a_s, a_d, emx, den, stream);
  final_add<<<((N * IN_F) + 255) / 256, 256, 0, stream>>>(C, b2, A, (float*)d_out,
                                                          N * IN_F);
}